// MANNNetwork_59476707115189
// MI455X (gfx1250) — compile-verified
//
#include <hip/hip_runtime.h>
#include <math.h>

// ---------------------------------------------------------------------------
// MANN network: input GEMM + gate GEMM via V_WMMA_F32_16X16X4_F32, then
// single-workgroup LDS-resident LSTM scan and MANN scan.
// Sizes (from reference): T=4096, D=512, F=256, H=200 (4H=800), K=40, N=128.
// ---------------------------------------------------------------------------

typedef __attribute__((ext_vector_type(2))) float v2f;
typedef __attribute__((ext_vector_type(8))) float v8f;

#define T_DIM 4096
#define D_DIM 512
#define F_DIM 256
#define HID   200
#define G4    800   // 4*HID
#define KD    40    // KEY_DIM
#define NS    128   // N_SLOTS
#define EPSV  1e-12f

__device__ __forceinline__ float sigmoidf_(float x) { return 1.0f / (1.0f + expf(-x)); }

// ---------------------------------------------------------------------------
// Kernel 1: xs[T,F] = x_train[T,D] @ W_in[D,F] + b_in[F]
// One wave per 16x16 output tile, K stepped by 4 via wmma_f32_16x16x4_f32.
// A layout (ISA 7.12.2, 32-bit A 16x4): lanes 0-15: vgpr0=K0, vgpr1=K1 at M=lane;
// lanes 16-31: vgpr0=K2, vgpr1=K3 at M=lane-16. B mirrored (row K striped over
// lanes as N). C/D: acc[r] = (M = r + 8*half, N = lane&15).
// ---------------------------------------------------------------------------
__global__ void gemm_xs_wmma(const float* __restrict__ X,
                             const float* __restrict__ W,
                             const float* __restrict__ bias,
                             float* __restrict__ out) {
    const int wavesPerBlock = blockDim.x >> 5;
    const int waveId = blockIdx.x * wavesPerBlock + (threadIdx.x >> 5);
    const int lane = threadIdx.x & 31;
    const int tilesN = F_DIM >> 4;                 // 16
    const int tileM = (waveId / tilesN) << 4;
    const int tileN = (waveId % tilesN) << 4;
    if (tileM >= T_DIM) return;

    const int half = lane >> 4;                    // 0: K0/K1, 1: K2/K3
    const int l16  = lane & 15;

    v8f acc = {};
    const float* arow = X + (size_t)(tileM + l16) * D_DIM + (half ? 2 : 0);
    const float* bcol = W + (size_t)(half ? 2 : 0) * F_DIM + tileN + l16;
    for (int k0 = 0; k0 < D_DIM; k0 += 4) {
        v2f a; a.x = arow[k0];            a.y = arow[k0 + 1];
        v2f b; b.x = bcol[(size_t)k0 * F_DIM];
               b.y = bcol[(size_t)(k0 + 1) * F_DIM];
        __builtin_prefetch(arow + k0 + 16, 0, 3);
        acc = __builtin_amdgcn_wmma_f32_16x16x4_f32(false, a, false, b,
                                                    (short)0, acc, false, false);
    }
    const float bn = bias[tileN + l16];
#pragma unroll
    for (int r = 0; r < 8; ++r) {
        const int row = tileM + r + (half ? 8 : 0);
        out[(size_t)row * F_DIM + tileN + l16] = acc[r] + bn;
    }
}

// ---------------------------------------------------------------------------
// Kernel 2: gx[T,4H] = [xs | y_shift] @ W_ih^T + b_ih + b_hh
// K=256 handled by WMMA; the single y_shift column (k=256) and biases fused in
// the epilogue. W_ih is [4H, F+1] row-major, so B[k][n] = W_ih[n*257 + k].
// ---------------------------------------------------------------------------
__global__ void gemm_gates_wmma(const float* __restrict__ XS,
                                const float* __restrict__ Wih,   // [800,257]
                                const float* __restrict__ bih,
                                const float* __restrict__ bhh,
                                const float* __restrict__ y,     // [T,1]
                                float* __restrict__ gx) {
    const int wavesPerBlock = blockDim.x >> 5;
    const int waveId = blockIdx.x * wavesPerBlock + (threadIdx.x >> 5);
    const int lane = threadIdx.x & 31;
    const int tilesN = G4 >> 4;                    // 50
    const int tileM = (waveId / tilesN) << 4;
    const int tileN = (waveId % tilesN) << 4;
    if (tileM >= T_DIM) return;

    const int half = lane >> 4;
    const int l16  = lane & 15;
    const int n    = tileN + l16;

    v8f acc = {};
    const float* arow = XS + (size_t)(tileM + l16) * F_DIM + (half ? 2 : 0);
    const float* brow = Wih + (size_t)n * (F_DIM + 1) + (half ? 2 : 0);
    for (int k0 = 0; k0 < F_DIM; k0 += 4) {
        v2f a; a.x = arow[k0];     a.y = arow[k0 + 1];
        v2f b; b.x = brow[k0];     b.y = brow[k0 + 1];
        acc = __builtin_amdgcn_wmma_f32_16x16x4_f32(false, a, false, b,
                                                    (short)0, acc, false, false);
    }
    const float wy   = Wih[(size_t)n * (F_DIM + 1) + F_DIM];  // k = 256 column
    const float bsum = bih[n] + bhh[n];
#pragma unroll
    for (int r = 0; r < 8; ++r) {
        const int row = tileM + r + (half ? 8 : 0);
        const float ysh = (row > 0) ? y[row - 1] : 0.0f;
        gx[(size_t)row * G4 + n] = acc[r] + ysh * wy + bsum;
    }
}

// ---------------------------------------------------------------------------
// Kernel 3: transpose W_hh [800,200] -> W_hhT [200,800] for coalesced reads
// ---------------------------------------------------------------------------
__global__ void transpose_whh(const float* __restrict__ Whh, float* __restrict__ WhhT) {
    int idx = blockIdx.x * blockDim.x + threadIdx.x;     // over 800*200
    if (idx >= G4 * HID) return;
    int j = idx / HID;      // gate row 0..799
    int k = idx % HID;      // hidden col 0..199
    WhhT[(size_t)k * G4 + j] = Whh[(size_t)j * HID + k];
}

// ---------------------------------------------------------------------------
// Kernel 4: sequential LSTM scan, one workgroup of 800 threads (25 waves).
// State h,c in LDS; per step each thread computes one gate pre-activation by
// a coalesced 200-MAC dot against W_hhT, then threads <200 update state and
// threads <41 produce keys / sigma.
// ---------------------------------------------------------------------------
__global__ void lstm_scan(const float* __restrict__ gx,     // [T,800]
                          const float* __restrict__ WhhT,   // [200,800]
                          const float* __restrict__ Wk,     // [200,40]
                          const float* __restrict__ bk,     // [40]
                          const float* __restrict__ Ws,     // [200,1]
                          const float* __restrict__ bs,     // [1]
                          float* __restrict__ keys,         // [T,40]
                          float* __restrict__ sigmas) {     // [T]
    __shared__ float h_lds[HID];
    __shared__ float c_lds[HID];
    __shared__ float g_lds[G4];
    const int j = threadIdx.x;    // 0..799
    if (j < HID) { h_lds[j] = 0.0f; c_lds[j] = 0.0f; }
    __syncthreads();

    for (int t = 0; t < T_DIM; ++t) {
        float gj = gx[(size_t)t * G4 + j];
        for (int k = 0; k < HID; ++k)
            gj = fmaf(h_lds[k], WhhT[(size_t)k * G4 + j], gj);
        g_lds[j] = gj;
        __syncthreads();

        if (j < HID) {
            const float ig = sigmoidf_(g_lds[j]);
            const float fg = sigmoidf_(g_lds[HID + j]);
            const float gg = tanhf(g_lds[2 * HID + j]);
            const float og = sigmoidf_(g_lds[3 * HID + j]);
            const float cn = fg * c_lds[j] + ig * gg;
            c_lds[j] = cn;
            h_lds[j] = og * tanhf(cn);
        }
        __syncthreads();

        if (j < KD) {
            float acc = bk[j];
            for (int k = 0; k < HID; ++k)
                acc = fmaf(h_lds[k], Wk[(size_t)k * KD + j], acc);
            keys[(size_t)t * KD + j] = tanhf(acc);
        } else if (j == KD) {
            float acc = bs[0];
            for (int k = 0; k < HID; ++k)
                acc = fmaf(h_lds[k], Ws[k], acc);
            sigmas[t] = sigmoidf_(acc);
        }
        __syncthreads();
    }
}

// ---------------------------------------------------------------------------
// Kernel 5: sequential MANN scan, one workgroup of 128 threads (one per slot).
// M kept in LDS with rows padded to 41 floats (gcd(41,64)=1 -> conflict-free).
// ---------------------------------------------------------------------------
#define MPAD 41
__global__ void mann_scan(const float* __restrict__ keys,    // [T,40]
                          const float* __restrict__ sigmas,  // [T]
                          const float* __restrict__ gammap,  // [1]
                          float* __restrict__ out) {         // [128,40]
    __shared__ float M[NS * MPAD];
    __shared__ float wr[NS];
    __shared__ float wu[NS];
    __shared__ float k_lds[KD];
    __shared__ float kn_lds[KD];
    __shared__ float red[NS];

    const int s = threadIdx.x;           // slot 0..127
    const float g = gammap[0];

    for (int d = 0; d < KD; ++d) M[s * MPAD + d] = 1e-6f;
    wr[s] = (s == 0) ? 1.0f : 0.0f;
    wu[s] = wr[s];
    __syncthreads();

    for (int t = 0; t < T_DIM; ++t) {
        if (s < KD) {
            const float kv = keys[(size_t)t * KD + s];
            k_lds[s]  = kv;
            kn_lds[s] = kv / fmaxf(fabsf(kv), EPSV);
        }
        const float sigma = sigmas[t];

        // max(wu) reduction
        red[s] = wu[s];
        __syncthreads();
        for (int off = NS >> 1; off > 0; off >>= 1) {
            if (s < off) red[s] = fmaxf(red[s], red[s + off]);
            __syncthreads();
        }
        const float maxwu = red[0];
        __syncthreads();

        const float wlu = (wu[s] <= maxwu) ? 1.0f : 0.0f;
        const float ww  = sigma * wr[s] + (1.0f - sigma) * wlu;

        // outer-product write + row norm + normalized dot
        float nrm2 = 0.0f;
        for (int d = 0; d < KD; ++d) {
            const float m = fmaf(ww, k_lds[d], M[s * MPAD + d]);
            M[s * MPAD + d] = m;
            nrm2 = fmaf(m, m, nrm2);
        }
        const float inv = 1.0f / fmaxf(sqrtf(nrm2), EPSV);
        float dot = 0.0f;
        for (int d = 0; d < KD; ++d)
            dot = fmaf(M[s * MPAD + d] * inv, kn_lds[d], dot);

        // softmax over slots
        red[s] = dot;
        __syncthreads();
        for (int off = NS >> 1; off > 0; off >>= 1) {
            if (s < off) red[s] = fmaxf(red[s], red[s + off]);
            __syncthreads();
        }
        const float dmax = red[0];
        __syncthreads();
        const float e = expf(dot - dmax);
        red[s] = e;
        __syncthreads();
        for (int off = NS >> 1; off > 0; off >>= 1) {
            if (s < off) red[s] += red[s + off];
            __syncthreads();
        }
        const float wrn = e / red[0];
        __syncthreads();

        wu[s] = g * wu[s] + wrn + ww;
        wr[s] = wrn;
        __syncthreads();
    }

    for (int d = 0; d < KD; ++d)
        out[(size_t)s * KD + d] = M[s * MPAD + d];
}

// ---------------------------------------------------------------------------
// Host-side launcher
// ---------------------------------------------------------------------------
extern "C" void kernel_launch(void* const* d_in, const int* in_sizes, int n_in,
                              void* d_out, int out_size, void* d_ws, size_t ws_size,
                              hipStream_t stream) {
    const float* x_train = (const float*)d_in[0];
    const float* y_train = (const float*)d_in[1];
    const float* W_in    = (const float*)d_in[2];
    const float* b_in    = (const float*)d_in[3];
    const float* W_ih    = (const float*)d_in[4];
    const float* W_hh    = (const float*)d_in[5];
    const float* b_ih    = (const float*)d_in[6];
    const float* b_hh    = (const float*)d_in[7];
    const float* W_k     = (const float*)d_in[8];
    const float* b_k     = (const float*)d_in[9];
    const float* W_s     = (const float*)d_in[10];
    const float* b_s     = (const float*)d_in[11];
    const float* gamma   = (const float*)d_in[12];
    float* out = (float*)d_out;

    float* ws = (float*)d_ws;
    float* xs     = ws;                                  // 4096*256
    float* gx     = xs + (size_t)T_DIM * F_DIM;          // 4096*800
    float* WhhT   = gx + (size_t)T_DIM * G4;             // 200*800
    float* keys   = WhhT + (size_t)HID * G4;             // 4096*40
    float* sigmas = keys + (size_t)T_DIM * KD;           // 4096

    // 1) xs = x_train @ W_in + b_in : 4096/16 * 256/16 = 4096 tiles, 4 waves/blk
    {
        const int tiles = (T_DIM / 16) * (F_DIM / 16);
        gemm_xs_wmma<<<tiles / 4, 128, 0, stream>>>(x_train, W_in, b_in, xs);
    }
    // 2) gx = [xs|y_shift] @ W_ih^T + b : 4096/16 * 800/16 = 12800 tiles
    {
        const int tiles = (T_DIM / 16) * (G4 / 16);
        gemm_gates_wmma<<<tiles / 4, 128, 0, stream>>>(xs, W_ih, b_ih, b_hh, y_train, gx);
    }
    // 3) transpose W_hh
    {
        const int n = G4 * HID;
        transpose_whh<<<(n + 255) / 256, 256, 0, stream>>>(W_hh, WhhT);
    }
    // 4) LSTM scan (one workgroup, 800 threads = 25 waves)
    lstm_scan<<<1, G4, 0, stream>>>(gx, WhhT, W_k, b_k, W_s, b_s, keys, sigmas);
    // 5) MANN scan (one workgroup, 128 threads)
    mann_scan<<<1, NS, 0, stream>>>(keys, sigmas, gamma, out);
}